// sLSTM_18124761989424
// MI455X (gfx1250) — compile-verified
//
#include <hip/hip_runtime.h>
#include <hip/hip_bf16.h>
#include <math.h>

#define B_ROWS 4096
#define D_DIM  1024
#define HN_    8
#define HD_    128
#define NGATE  4096          // 4 gates * 1024
#define PROJ_  1365
#define UPN    2730          // 2 * PROJ
#define UPN_PAD 2816         // 22 * 128 (padded rows for up weights)
#define KPAD   1376          // PROJ padded to multiple of 32 (43*32)
#define XHK    2048          // [x | h] concatenated activation row
#define WCATK  1152          // [W row (1024) | R row (128)] concatenated weight row
#define EPS_   1e-5f

typedef __attribute__((ext_vector_type(16))) __bf16 v16bf;
typedef __attribute__((ext_vector_type(8)))  float  v8f;

#define BM 128
#define BN 128
#define BK 32
#define LDSS 40              // padded LDS row stride (elements) to spread banks
#define THREADS 256

// ---------- helpers ----------

// Pack two f32 -> packed bf16 pair: round-bias add + one v_perm_b32.
__device__ __forceinline__ unsigned int pack2(float lo, float hi) {
  union { float f; unsigned int u; } a, b;
  a.f = lo; b.f = hi;
  return __builtin_amdgcn_perm(b.u + 0x8000u, a.u + 0x8000u, 0x07060302u);
}

union FragU { uint4 q[2]; v16bf v; };

// Per-lane 16-bit WMMA fragment per ISA §7.12.2: row/col = lane&15,
// chunk1 = K[kb..kb+7], chunk2 = K[kb+16..kb+23], kb = (lane>>4)*8.
__device__ __forceinline__ v16bf load_frag(const unsigned short* rowBase, int kb) {
  FragU f;
  f.q[0] = *(const uint4*)(rowBase + kb);
  f.q[1] = *(const uint4*)(rowBase + 16 + kb);
  return f.v;
}

// Async copy of one thread's 32-byte tile slice: global -> LDS, no VGPR staging.
// INST_OFFSET is added to both the LDS and global addresses (ISA §10.7.4).
__device__ __forceinline__ void async_copy32(unsigned short* dstLds,
                                             const unsigned short* srcGlb) {
  unsigned la = (unsigned)(unsigned long long)dstLds;   // LDS aperture: addr[31:0]
  asm volatile("global_load_async_to_lds_b128 %0, %1, off"
               :: "v"(la), "v"(srcGlb) : "memory");
  asm volatile("global_load_async_to_lds_b128 %0, %1, off offset:16"
               :: "v"(la), "v"(srcGlb) : "memory");
}

__device__ __forceinline__ void wait_async4() {
  asm volatile("s_wait_asynccnt 0x4" ::: "memory");
}

// Tied-accumulator WMMA: D == C forced into the same 8-VGPR tuple -> no PHI copies.
#define WMMA_BF16(accv, av, bv)                                         \
  asm volatile("v_wmma_f32_16x16x32_bf16 %0, %1, %2, %0"                \
               : "+v"(accv) : "v"(av), "v"(bv))

// One BK=32 step: each wave computes a 32x64 C-tile = 8 WMMAs.
__device__ __forceinline__ void wave_mma(const unsigned short* sA, const unsigned short* sW,
                                         int wm, int wn, int lane, v8f acc[2][4]) {
  int rh = lane & 15;
  int kb = (lane >> 4) * 8;
  v16bf a0 = load_frag(sA + (wm * 32 +      rh) * LDSS, kb);
  v16bf a1 = load_frag(sA + (wm * 32 + 16 + rh) * LDSS, kb);
#pragma unroll
  for (int j = 0; j < 4; ++j) {
    v16bf bfr = load_frag(sW + (wn * 64 + j * 16 + rh) * LDSS, kb);
    WMMA_BF16(acc[0][j], a0, bfr);
    WMMA_BF16(acc[1][j], a1, bfr);
  }
}

// WMMA -> VALU hazard spacer (4 co-exec slots; inline asm bypasses compiler hazards).
__device__ __forceinline__ void wmma_valu_fence() {
  asm volatile("s_nop 4" :::);
}

__device__ __forceinline__ void zero_acc(v8f acc[2][4]) {
#pragma unroll
  for (int i = 0; i < 2; ++i)
#pragma unroll
    for (int j = 0; j < 4; ++j)
#pragma unroll
      for (int r = 0; r < 8; ++r) acc[i][j][r] = 0.0f;
}

// ---------- kernel 0a: generic f32 -> bf16 image conversion (zero padded) ----------

__global__ __launch_bounds__(256)
void cvt_bf16_kernel(const float* __restrict__ src, unsigned short* __restrict__ dst,
                     int Ksrc, int Kdst, int Nsrc, int npack) {
  int p = blockIdx.x * blockDim.x + threadIdx.x;
  if (p >= npack) return;
  int row  = (p * 8) / Kdst;
  int col0 = (p * 8) % Kdst;
  uint4 o;
  if (Ksrc == Kdst && row < Nsrc) {
    const float* s = src + (size_t)row * Ksrc + col0;   // 16B aligned (Ksrc%4==0, col0%8==0)
    float4 v0 = *(const float4*)(s);
    float4 v1 = *(const float4*)(s + 4);
    o.x = pack2(v0.x, v0.y); o.y = pack2(v0.z, v0.w);
    o.z = pack2(v1.x, v1.y); o.w = pack2(v1.z, v1.w);
  } else {
    float v[8];
#pragma unroll
    for (int j = 0; j < 8; ++j) {
      int k = col0 + j;
      v[j] = (row < Nsrc && k < Ksrc) ? src[(size_t)row * Ksrc + k] : 0.0f;
    }
    o.x = pack2(v[0], v[1]); o.y = pack2(v[2], v[3]);
    o.z = pack2(v[4], v[5]); o.w = pack2(v[6], v[7]);
  }
  *(uint4*)(dst + (size_t)p * 8) = o;
}

// ---------- kernel 0b: build concatenated gates weight image Wcat[4096][1152] ----------
// row n: cols 0..1023 = Wgate[n%1024][:], cols 1024..1151 = R[gate][head][n&127][:]

__global__ __launch_bounds__(256)
void build_wcat_kernel(const float* __restrict__ Wi, const float* __restrict__ Wf,
                       const float* __restrict__ Wz, const float* __restrict__ Wo,
                       const float* __restrict__ Ri, const float* __restrict__ Rf,
                       const float* __restrict__ Rz, const float* __restrict__ Ro,
                       unsigned short* __restrict__ dst) {
  int n = blockIdx.x;               // 0..4095
  int t = threadIdx.x;
  int gate = n >> 10, colg = n & 1023, head = colg >> 7, e = colg & 127;
  const float* W = (gate == 0) ? Wi : (gate == 1) ? Wf : (gate == 2) ? Wz : Wo;
  const float* R = ((gate == 0) ? Ri : (gate == 1) ? Rf : (gate == 2) ? Rz : Ro)
                   + (size_t)(head * HD_ + e) * HD_;
  unsigned short* drow = dst + (size_t)n * WCATK;
  int c0 = 4 * t;
  float4 w4 = *(const float4*)(W + (size_t)colg * D_DIM + c0);
  uint2 p; p.x = pack2(w4.x, w4.y); p.y = pack2(w4.z, w4.w);
  *(uint2*)(&drow[c0]) = p;
  if (t < HD_ / 4) {
    float4 r4 = *(const float4*)(R + 4 * t);
    uint2 q; q.x = pack2(r4.x, r4.y); q.y = pack2(r4.z, r4.w);
    *(uint2*)(&drow[D_DIM + 4 * t]) = q;
  }
}

// ---------- kernel 1: LayerNorm(seq) -> xh[:,0:1024] ; h_tm1 -> xh[:,1024:2048] ----------

__global__ __launch_bounds__(256)
void ln_kernel(const float* __restrict__ seq, const float* __restrict__ h_tm1,
               const float* __restrict__ ln_w, const float* __restrict__ ln_b,
               unsigned short* __restrict__ xh) {
  int b = blockIdx.x;
  int t = threadIdx.x;
  size_t base = (size_t)b * D_DIM;
  int c0 = t * 4;
  float4 x4 = *(const float4*)(seq + base + c0);
  float4 h4 = *(const float4*)(h_tm1 + base + c0);
  float s  = x4.x + x4.y + x4.z + x4.w;
  float s2 = x4.x * x4.x + x4.y * x4.y + x4.z * x4.z + x4.w * x4.w;

  __shared__ float red[2][8];
  for (int off = 16; off; off >>= 1) {
    s  += __shfl_xor(s,  off, 32);
    s2 += __shfl_xor(s2, off, 32);
  }
  int wave = t >> 5, lane = t & 31;
  if (lane == 0) { red[0][wave] = s; red[1][wave] = s2; }
  __syncthreads();
  if (wave == 0) {
    float a  = (lane < 8) ? red[0][lane] : 0.f;
    float b2 = (lane < 8) ? red[1][lane] : 0.f;
    for (int off = 4; off; off >>= 1) { a += __shfl_xor(a, off, 32); b2 += __shfl_xor(b2, off, 32); }
    if (lane == 0) { red[0][0] = a; red[1][0] = b2; }
  }
  __syncthreads();
  float mean = red[0][0] * (1.0f / D_DIM);
  float var  = red[1][0] * (1.0f / D_DIM) - mean * mean;
  float rstd = rsqrtf(var + EPS_);

  float4 w4 = *(const float4*)(ln_w + c0);
  float4 b4 = *(const float4*)(ln_b + c0);
  float n0 = (x4.x - mean) * rstd * w4.x + b4.x;
  float n1 = (x4.y - mean) * rstd * w4.y + b4.y;
  float n2 = (x4.z - mean) * rstd * w4.z + b4.z;
  float n3 = (x4.w - mean) * rstd * w4.w + b4.w;
  uint2 px; px.x = pack2(n0, n1); px.y = pack2(n2, n3);
  uint2 ph; ph.x = pack2(h4.x, h4.y); ph.y = pack2(h4.z, h4.w);
  unsigned short* row = xh + (size_t)b * XHK;
  *(uint2*)(&row[c0])         = px;
  *(uint2*)(&row[D_DIM + c0]) = ph;
}

// ---------- kernel 2: fused gates GEMM (uniform K=1152, async double-buffered) ----------

__global__ __launch_bounds__(THREADS)
void gates_gemm_kernel(const unsigned short* __restrict__ xh,     // [4096][2048]
                       const unsigned short* __restrict__ Wcat,   // [4096][1152]
                       const float* __restrict__ bi, const float* __restrict__ bff,
                       const float* __restrict__ bz, const float* __restrict__ bo,
                       float* __restrict__ G) {
  __shared__ __align__(16) unsigned short sA[2][BM * LDSS];
  __shared__ __align__(16) unsigned short sW[2][BN * LDSS];
  int tid = threadIdx.x, lane = tid & 31, wave = tid >> 5;
  int wm = wave & 3, wn = wave >> 2;
  int n0 = blockIdx.x * BN;          // 0..4095
  int m0 = blockIdx.y * BM;
  int gate = n0 >> 10;
  int colg = n0 & 1023;              // column within gate
  int head = colg >> 7;
  const float* bias = (gate == 0) ? bi : (gate == 1) ? bff : (gate == 2) ? bz : bo;
  int hoff = head * HD_;             // extra A-col offset during recurrent phase

  v8f acc[2][4];
  zero_acc(acc);

  const int STEPS_D = D_DIM / BK;    // 32
  const int STEPS   = WCATK / BK;    // 36

  int r = tid >> 1, c = (tid & 1) * 16;
  const unsigned short* gA = xh   + (size_t)(m0 + r) * XHK   + c;
  const unsigned short* gW = Wcat + (size_t)(n0 + r) * WCATK + c;
  unsigned short* lA = &sA[0][0] + r * LDSS + c;
  unsigned short* lW = &sW[0][0] + r * LDSS + c;
  const int BUF = BM * LDSS;         // elements per buffer

  async_copy32(lA, gA);
  async_copy32(lW, gW);
  for (int step = 0; step < STEPS; ++step) {
    int nb = (step + 1) & 1;
    int ns = step + 1 < STEPS ? step + 1 : step;        // clamp (cndmask)
    int kA = ns * BK + (ns >= STEPS_D ? hoff : 0);      // cndmask
    async_copy32(lA + nb * BUF, gA + kA);
    async_copy32(lW + nb * BUF, gW + ns * BK);
    wait_async4();
    __syncthreads();
    wave_mma(sA[step & 1], sW[step & 1], wm, wn, lane, acc);
    __syncthreads();
  }
  wmma_valu_fence();

  int rowHalf = (lane >> 4) * 8;
#pragma unroll
  for (int i = 0; i < 2; ++i) {
#pragma unroll
    for (int j = 0; j < 4; ++j) {
      int colT = wn * 64 + j * 16 + (lane & 15);
      float bv = bias[colg + colT];
#pragma unroll
      for (int r2 = 0; r2 < 8; ++r2) {
        int row = m0 + wm * 32 + i * 16 + rowHalf + r2;
        G[(size_t)row * NGATE + n0 + colT] = acc[i][j][r2] + bv;
      }
    }
  }
}

// ---------- kernel 3: sLSTM cell + GroupNorm (1 wave per head) ----------

__global__ __launch_bounds__(256)
void cell_kernel(const float* __restrict__ G,
                 const float* __restrict__ c_tm1, const float* __restrict__ n_tm1,
                 const float* __restrict__ m_tm1,
                 const float* __restrict__ gn_w, const float* __restrict__ gn_b,
                 float* __restrict__ outAll, unsigned short* __restrict__ gout_bf) {
  int b = blockIdx.x, t = threadIdx.x;
  const float* g = G + (size_t)b * NGATE;
  size_t base = (size_t)b * D_DIM;
  const size_t S = (size_t)B_ROWS * D_DIM;
  int c0 = t * 4;

  float4 it4 = *(const float4*)(g + c0);
  float4 ft4 = *(const float4*)(g + 1024 + c0);
  float4 zt4 = *(const float4*)(g + 2048 + c0);
  float4 ot4 = *(const float4*)(g + 3072 + c0);
  float4 cp4 = *(const float4*)(c_tm1 + base + c0);
  float4 np4 = *(const float4*)(n_tm1 + base + c0);
  float4 mp4 = *(const float4*)(m_tm1 + base + c0);

  float it[4] = {it4.x, it4.y, it4.z, it4.w};
  float ft[4] = {ft4.x, ft4.y, ft4.z, ft4.w};
  float zt[4] = {zt4.x, zt4.y, zt4.z, zt4.w};
  float ot[4] = {ot4.x, ot4.y, ot4.z, ot4.w};
  float cp[4] = {cp4.x, cp4.y, cp4.z, cp4.w};
  float np[4] = {np4.x, np4.y, np4.z, np4.w};
  float mp[4] = {mp4.x, mp4.y, mp4.z, mp4.w};

  float ct[4], nt[4], ht[4], mt[4];
  float s = 0.f, s2 = 0.f;
#pragma unroll
  for (int i = 0; i < 4; ++i) {
    mt[i] = fmaxf(ft[i] + mp[i], it[i]);
    float ie = expf(it[i] - mt[i]);
    float fe = expf(ft[i] - mt[i] + mp[i]);
    float za = tanhf(zt[i]);
    float oa = 1.0f / (1.0f + expf(-ot[i]));
    ct[i] = fe * cp[i] + ie * za;
    nt[i] = fe * np[i] + ie;
    ht[i] = oa * (ct[i] / nt[i]);
    s += ht[i]; s2 += ht[i] * ht[i];
  }
  *(float4*)(outAll + S * 1 + base + c0) = make_float4(ct[0], ct[1], ct[2], ct[3]);
  *(float4*)(outAll + S * 2 + base + c0) = make_float4(nt[0], nt[1], nt[2], nt[3]);
  *(float4*)(outAll + S * 3 + base + c0) = make_float4(ht[0], ht[1], ht[2], ht[3]);
  *(float4*)(outAll + S * 4 + base + c0) = make_float4(mt[0], mt[1], mt[2], mt[3]);

  // wave32 covers exactly one head (32 lanes * 4 contiguous cols = 128)
  for (int off = 16; off; off >>= 1) {
    s  += __shfl_xor(s,  off, 32);
    s2 += __shfl_xor(s2, off, 32);
  }
  float mean = s * (1.0f / HD_);
  float var  = s2 * (1.0f / HD_) - mean * mean;
  float rstd = rsqrtf(var + EPS_);

  float4 w4 = *(const float4*)(gn_w + c0);
  float4 b4 = *(const float4*)(gn_b + c0);
  float g0 = (ht[0] - mean) * rstd * w4.x + b4.x;
  float g1 = (ht[1] - mean) * rstd * w4.y + b4.y;
  float g2 = (ht[2] - mean) * rstd * w4.z + b4.z;
  float g3 = (ht[3] - mean) * rstd * w4.w + b4.w;
  uint2 p; p.x = pack2(g0, g1); p.y = pack2(g2, g3);
  *(uint2*)(&gout_bf[base + c0]) = p;
}

// ---------- kernel 4: up GEMM  up = gout @ up_w^T + up_b (async double-buffered) ----------

__global__ __launch_bounds__(THREADS)
void up_gemm_kernel(const unsigned short* __restrict__ A,
                    const unsigned short* __restrict__ Wb,   // [2816][1024], zero-padded rows
                    const float* __restrict__ bias, float* __restrict__ C) {
  __shared__ __align__(16) unsigned short sA[2][BM * LDSS];
  __shared__ __align__(16) unsigned short sW[2][BN * LDSS];
  int tid = threadIdx.x, lane = tid & 31, wave = tid >> 5;
  int wm = wave & 3, wn = wave >> 2;
  int n0 = blockIdx.x * BN;
  int m0 = blockIdx.y * BM;

  v8f acc[2][4];
  zero_acc(acc);

  const int STEPS = D_DIM / BK;       // 32
  int r = tid >> 1, c = (tid & 1) * 16;
  const unsigned short* gA = A  + (size_t)(m0 + r) * D_DIM + c;
  const unsigned short* gW = Wb + (size_t)(n0 + r) * D_DIM + c;
  unsigned short* lA = &sA[0][0] + r * LDSS + c;
  unsigned short* lW = &sW[0][0] + r * LDSS + c;
  const int BUF = BM * LDSS;

  async_copy32(lA, gA);
  async_copy32(lW, gW);
  for (int step = 0; step < STEPS; ++step) {
    int nb = (step + 1) & 1;
    int ns = step + 1 < STEPS ? step + 1 : step;   // clamp (cndmask)
    async_copy32(lA + nb * BUF, gA + ns * BK);
    async_copy32(lW + nb * BUF, gW + ns * BK);
    wait_async4();
    __syncthreads();
    wave_mma(sA[step & 1], sW[step & 1], wm, wn, lane, acc);
    __syncthreads();
  }
  wmma_valu_fence();

  int rowHalf = (lane >> 4) * 8;
#pragma unroll
  for (int i = 0; i < 2; ++i)
#pragma unroll
    for (int j = 0; j < 4; ++j) {
      int n = n0 + wn * 64 + j * 16 + (lane & 15);
      if (n < UPN) {
        float bv = bias[n];
#pragma unroll
        for (int r2 = 0; r2 < 8; ++r2) {
          int row = m0 + wm * 32 + i * 16 + rowHalf + r2;
          C[(size_t)row * UPN + n] = acc[i][j][r2] + bv;
        }
      }
    }
}

// ---------- kernel 5: GLU with exact GELU -> bf16 padded activations ----------

__global__ __launch_bounds__(256)
void glu_kernel(const float* __restrict__ up, unsigned short* __restrict__ act) {
  int b = blockIdx.x;
  const float* rowA = up + (size_t)b * UPN;
  const float* rowX = rowA + PROJ_;
  unsigned int* dst = (unsigned int*)(act + (size_t)b * KPAD);
  for (int p = threadIdx.x; p < KPAD / 2; p += blockDim.x) {
    int j0 = 2 * p, j1 = 2 * p + 1;
    float v0 = 0.f, v1 = 0.f;
    if (j0 < PROJ_) {
      float x0 = rowX[j0];
      v0 = rowA[j0] + 0.5f * x0 * (1.0f + erff(x0 * 0.70710678118654752f));
    }
    if (j1 < PROJ_) {
      float x1 = rowX[j1];
      v1 = rowA[j1] + 0.5f * x1 * (1.0f + erff(x1 * 0.70710678118654752f));
    }
    dst[p] = pack2(v0, v1);
  }
}

// ---------- kernel 6: down GEMM + residual (async double-buffered) ----------

__global__ __launch_bounds__(THREADS)
void down_gemm_kernel(const unsigned short* __restrict__ A,      // [4096][1376]
                      const unsigned short* __restrict__ Wb,     // [1024][1376], zero-padded K
                      const float* __restrict__ bias, const float* __restrict__ seq,
                      float* __restrict__ C) {
  __shared__ __align__(16) unsigned short sA[2][BM * LDSS];
  __shared__ __align__(16) unsigned short sW[2][BN * LDSS];
  int tid = threadIdx.x, lane = tid & 31, wave = tid >> 5;
  int wm = wave & 3, wn = wave >> 2;
  int n0 = blockIdx.x * BN;
  int m0 = blockIdx.y * BM;

  v8f acc[2][4];
  zero_acc(acc);

  const int STEPS = KPAD / BK;        // 43
  int r = tid >> 1, c = (tid & 1) * 16;
  const unsigned short* gA = A  + (size_t)(m0 + r) * KPAD + c;
  const unsigned short* gW = Wb + (size_t)(n0 + r) * KPAD + c;
  unsigned short* lA = &sA[0][0] + r * LDSS + c;
  unsigned short* lW = &sW[0][0] + r * LDSS + c;
  const int BUF = BM * LDSS;

  async_copy32(lA, gA);
  async_copy32(lW, gW);
  for (int step = 0; step < STEPS; ++step) {
    int nb = (step + 1) & 1;
    int ns = step + 1 < STEPS ? step + 1 : step;   // clamp (cndmask)
    async_copy32(lA + nb * BUF, gA + ns * BK);
    async_copy32(lW + nb * BUF, gW + ns * BK);
    wait_async4();
    __syncthreads();
    wave_mma(sA[step & 1], sW[step & 1], wm, wn, lane, acc);
    __syncthreads();
  }
  wmma_valu_fence();

  int rowHalf = (lane >> 4) * 8;
#pragma unroll
  for (int i = 0; i < 2; ++i)
#pragma unroll
    for (int j = 0; j < 4; ++j) {
      int n = n0 + wn * 64 + j * 16 + (lane & 15);
      float bv = bias[n];
#pragma unroll
      for (int r2 = 0; r2 < 8; ++r2) {
        int row = m0 + wm * 32 + i * 16 + rowHalf + r2;
        C[(size_t)row * D_DIM + n] = acc[i][j][r2] + bv + seq[(size_t)row * D_DIM + n];
      }
    }
}

// ---------- host launcher ----------

extern "C" void kernel_launch(void* const* d_in, const int* in_sizes, int n_in,
                              void* d_out, int out_size, void* d_ws, size_t ws_size,
                              hipStream_t stream) {
  const float* seq    = (const float*)d_in[0];
  const float* c_tm1  = (const float*)d_in[1];
  const float* n_tm1  = (const float*)d_in[2];
  const float* h_tm1  = (const float*)d_in[3];
  const float* m_tm1  = (const float*)d_in[4];
  const float* ln_w   = (const float*)d_in[5];
  const float* ln_b   = (const float*)d_in[6];
  const float* gn_w   = (const float*)d_in[7];
  const float* gn_b   = (const float*)d_in[8];
  const float* Wi_w   = (const float*)d_in[9];
  const float* Wi_b   = (const float*)d_in[10];
  const float* Wf_w   = (const float*)d_in[11];
  const float* Wf_b   = (const float*)d_in[12];
  const float* Wz_w   = (const float*)d_in[13];
  const float* Wz_b   = (const float*)d_in[14];
  const float* Wo_w   = (const float*)d_in[15];
  const float* Wo_b   = (const float*)d_in[16];
  const float* Ri     = (const float*)d_in[17];
  const float* Rf     = (const float*)d_in[18];
  const float* Rz     = (const float*)d_in[19];
  const float* Ro     = (const float*)d_in[20];
  const float* up_w   = (const float*)d_in[21];
  const float* up_b   = (const float*)d_in[22];
  const float* down_w = (const float*)d_in[23];
  const float* down_b = (const float*)d_in[24];
  float* outp = (float*)d_out;

  char* ws = (char*)d_ws;
  size_t off = 0;
  unsigned short* xh_bf   = (unsigned short*)(ws + off); off += (size_t)B_ROWS * XHK * 2;
  unsigned short* gout_bf = (unsigned short*)(ws + off); off += (size_t)B_ROWS * D_DIM * 2;
  unsigned short* act_bf  = (unsigned short*)(ws + off); off += (size_t)B_ROWS * KPAD * 2;
  unsigned short* Wcat_bf = (unsigned short*)(ws + off); off += (size_t)NGATE * WCATK * 2;
  unsigned short* up_bf   = (unsigned short*)(ws + off); off += (size_t)UPN_PAD * D_DIM * 2;
  unsigned short* down_bf = (unsigned short*)(ws + off); off += (size_t)D_DIM * KPAD * 2;
  // gates buffer (64 MB) and up buffer (44.7 MB) have disjoint lifetimes -> alias
  float* G  = (float*)(ws + off);
  float* up = (float*)(ws + off);

  // --- one-shot weight image builds (f32 -> bf16, zero padded) ---
  build_wcat_kernel<<<NGATE, 256, 0, stream>>>(Wi_w, Wf_w, Wz_w, Wo_w,
                                               Ri, Rf, Rz, Ro, Wcat_bf);
  {
    int np = UPN_PAD * D_DIM / 8;
    cvt_bf16_kernel<<<(np + 255) / 256, 256, 0, stream>>>(
        up_w, up_bf, D_DIM, D_DIM, UPN, np);
    int npd = D_DIM * KPAD / 8;
    cvt_bf16_kernel<<<(npd + 255) / 256, 256, 0, stream>>>(
        down_w, down_bf, PROJ_, KPAD, D_DIM, npd);
  }

  ln_kernel<<<B_ROWS, 256, 0, stream>>>(seq, h_tm1, ln_w, ln_b, xh_bf);

  gates_gemm_kernel<<<dim3(NGATE / BN, B_ROWS / BM), THREADS, 0, stream>>>(
      xh_bf, Wcat_bf, Wi_b, Wf_b, Wz_b, Wo_b, G);

  cell_kernel<<<B_ROWS, 256, 0, stream>>>(G, c_tm1, n_tm1, m_tm1, gn_w, gn_b,
                                          outp, gout_bf);

  up_gemm_kernel<<<dim3(UPN_PAD / BN, B_ROWS / BM), THREADS, 0, stream>>>(
      gout_bf, up_bf, up_b, up);

  glu_kernel<<<B_ROWS, 256, 0, stream>>>(up, act_bf);

  down_gemm_kernel<<<dim3(D_DIM / BN, B_ROWS / BM), THREADS, 0, stream>>>(
      act_bf, down_bf, down_b, seq, outp);
}